// CTPN_Loss_25177098289550
// MI455X (gfx1250) — compile-verified
//
#include <hip/hip_runtime.h>

// ---------------------------------------------------------------------------
// CTPN loss on MI455X (gfx1250, wave32).
// Latency-bound gather-reduction; WMMA f32 16x16x4 (f32 in / f32 accum — the
// only precision-safe matrix op for a loss sum) used as a one-instruction
// 32-lane reduction tree. Deterministic: per-block partials + fixed-order
// final sum (no float atomics).
// ---------------------------------------------------------------------------

typedef __attribute__((ext_vector_type(2))) float v2f;
typedef __attribute__((ext_vector_type(8))) float v8f;

#define HH      128
#define WWW     192
#define KCH     10
#define PLANE   (HH * WWW)        // per-channel stride = 24576
#define NPOS    64
#define NNEG    64
#define NV      20000
#define NO      5000
#define NITEMS  (NPOS + NNEG + NV + NO)   // 25128
#define NBLOCKS 32
#define NTHREADS 256
#define NWAVES  (NTHREADS / 32)

__device__ __forceinline__ float smooth_l1(float d) {
    float ad = fabsf(d);
    return (ad < 1.0f) ? (0.5f * ad * ad) : (ad - 0.5f);
}

// Sum a float across the 32 lanes of a wave using V_WMMA_F32_16X16X4_F32.
// A (16x4 f32, 2 VGPRs/lane): a.x = p, a.y = 0 -> each lane's p occupies
// exactly one A slot regardless of exact K placement. B = all ones (layout
// independent). D[m][n] = p_m + p_{m+16} for every column n, so with the
// documented C/D layout (lanes 0-15: rows 0-7, lanes 16-31: rows 8-15):
//   sum(d[0..7]) in lane 0  = sum_{m=0..7}  (p_m + p_{m+16})
//   sum(d[0..7]) in lane 16 = sum_{m=8..15} (p_m + p_{m+16})
// total = lane0 + lane16.
__device__ __forceinline__ float wave_sum_f32(float p) {
    v2f a; a.x = p;    a.y = 0.0f;
    v2f b; b.x = 1.0f; b.y = 1.0f;
    v8f c = {};
    v8f d = __builtin_amdgcn_wmma_f32_16x16x4_f32(
        /*neg_a=*/false, a, /*neg_b=*/false, b,
        /*c_mod=*/(short)0, c, /*reuse_a=*/false, /*reuse_b=*/false);
    float s = d[0] + d[1] + d[2] + d[3] + d[4] + d[5] + d[6] + d[7];
    return __shfl(s, 0, 32) + __shfl(s, 16, 32);
}

__global__ void ctpn_loss_main(const float* __restrict__ score,
                               const float* __restrict__ vpred,
                               const float* __restrict__ side,
                               const int*   __restrict__ pos,
                               const int*   __restrict__ neg,
                               const int*   __restrict__ vidx,
                               const float* __restrict__ vtgt,
                               const int*   __restrict__ sidx,
                               const float* __restrict__ stgt,
                               float*       __restrict__ partials) {
    float pc = 0.0f;   // cls CE sum
    float pv = 0.0f;   // vertical smooth-L1 sum (already /2 per item)
    float po = 0.0f;   // side smooth-L1 sum

    int tid    = blockIdx.x * blockDim.x + threadIdx.x;
    int stride = gridDim.x * blockDim.x;

    for (int i = tid; i < NITEMS; i += stride) {
        if (i < NPOS + NNEG) {
            // -------- classification term --------
            bool is_pos = (i < NPOS);
            const int* ip = is_pos ? (pos + 3 * i) : (neg + 3 * (i - NPOS));
            int x = ip[0], y = ip[1], a = ip[2];
            int base = ((2 * a) * HH + y) * WWW + x;
            float l0 = score[base];
            float l1 = score[base + PLANE];
            float m  = fmaxf(l0, l1);
            float lse = m + logf(__expf(l0 - m) + __expf(l1 - m));
            pc += lse - (is_pos ? l1 : l0);
        } else if (i < NPOS + NNEG + NV) {
            // -------- vertical regression term --------
            int j = i - (NPOS + NNEG);
            int x = vidx[3 * j], y = vidx[3 * j + 1], a = vidx[3 * j + 2];
            int base = ((2 * a) * HH + y) * WWW + x;
            float d0 = vpred[base]         - vtgt[2 * j];
            float d1 = vpred[base + PLANE] - vtgt[2 * j + 1];
            pv += 0.5f * (smooth_l1(d0) + smooth_l1(d1));   // mean over axis -1
        } else {
            // -------- side refinement term --------
            int j = i - (NPOS + NNEG + NV);
            int x = sidx[3 * j], y = sidx[3 * j + 1], a = sidx[3 * j + 2];
            float sp = side[(a * HH + y) * WWW + x];
            po += smooth_l1(sp - stgt[j]);
        }
    }

    // -------- block reduction: WMMA per wave, LDS across waves --------
    // All threads reconverge here (structured loop exit) -> EXEC all 1s,
    // as required by WMMA.
    float wc = wave_sum_f32(pc);
    float wv = wave_sum_f32(pv);
    float wo = wave_sum_f32(po);

    __shared__ float lds[NWAVES][3];
    int wave = threadIdx.x >> 5;
    int lane = threadIdx.x & 31;
    if (lane == 0) {
        lds[wave][0] = wc;
        lds[wave][1] = wv;
        lds[wave][2] = wo;
    }
    __syncthreads();

    if (threadIdx.x == 0) {
        float bc = 0.0f, bv = 0.0f, bo = 0.0f;
        #pragma unroll
        for (int w = 0; w < NWAVES; ++w) {
            bc += lds[w][0];
            bv += lds[w][1];
            bo += lds[w][2];
        }
        partials[3 * blockIdx.x + 0] = bc;
        partials[3 * blockIdx.x + 1] = bv;
        partials[3 * blockIdx.x + 2] = bo;
    }
}

__global__ void ctpn_loss_final(const float* __restrict__ partials,
                                float* __restrict__ out) {
    if (blockIdx.x == 0 && threadIdx.x == 0) {
        float c = 0.0f, v = 0.0f, o = 0.0f;
        for (int b = 0; b < NBLOCKS; ++b) {   // fixed order -> deterministic
            c += partials[3 * b + 0];
            v += partials[3 * b + 1];
            o += partials[3 * b + 2];
        }
        float cls_loss   = c / 128.0f;          // NS
        float v_reg_loss = v / (float)NV;
        float o_reg_loss = o / (float)NO;
        out[0] = cls_loss + v_reg_loss + o_reg_loss;
        out[1] = cls_loss;
        out[2] = v_reg_loss;
        out[3] = o_reg_loss;
    }
}

extern "C" void kernel_launch(void* const* d_in, const int* in_sizes, int n_in,
                              void* d_out, int out_size, void* d_ws, size_t ws_size,
                              hipStream_t stream) {
    const float* score = (const float*)d_in[0];  // (1, 2K, H, W)
    const float* vpred = (const float*)d_in[1];  // (1, 2K, H, W)
    const float* side  = (const float*)d_in[2];  // (1, K, H, W)
    const int*   pos   = (const int*)  d_in[3];  // (64, 3)
    const int*   neg   = (const int*)  d_in[4];  // (64, 3)
    const int*   vidx  = (const int*)  d_in[5];  // (20000, 3)
    const float* vtgt  = (const float*)d_in[6];  // (20000, 2)
    const int*   sidx  = (const int*)  d_in[7];  // (5000, 3)
    const float* stgt  = (const float*)d_in[8];  // (5000,)

    float* partials = (float*)d_ws;              // NBLOCKS*3 floats, fully overwritten
    float* out      = (float*)d_out;             // 4 floats

    ctpn_loss_main<<<NBLOCKS, NTHREADS, 0, stream>>>(
        score, vpred, side, pos, neg, vidx, vtgt, sidx, stgt, partials);
    ctpn_loss_final<<<1, 1, 0, stream>>>(partials, out);
}